// CombinedLabelDistributionLoss_7791070674972
// MI455X (gfx1250) — compile-verified
//
#include <hip/hip_runtime.h>
#include <math.h>

// ---------------- problem geometry ----------------
#define T_LEN   4096
#define B_ROWS  64
#define N_TOT   (B_ROWS * T_LEN)      // 262144
#define NFREQ   140
#define NFREQ_PAD 144                 // 9 tiles of 16
#define FTILES  9
#define N2      64                    // second DFT factor (N = 4096 * 64)
#define KSEG    16                    // split-K segments (deterministic)
#define KPER    (T_LEN / KSEG)        // 256 K-units per segment
#define TILE_ELEMS (NFREQ_PAD * N2)   // 9216

typedef __attribute__((ext_vector_type(2))) float v2f;
typedef __attribute__((ext_vector_type(8))) float v8f;

// ---------------------------------------------------------------------------
// Kernel 1: per-row standardization stats + Pearson + windowed signal
// ---------------------------------------------------------------------------
__global__ __launch_bounds__(256)
void k_rowstats(const float* __restrict__ pred, const float* __restrict__ targ,
                float* __restrict__ sig, float* __restrict__ pearson)
{
    const int row = blockIdx.x;
    const int tid = threadIdx.x;
    const float* x = pred + row * T_LEN;
    const float* y = targ + row * T_LEN;

    float sx = 0.f, sxx = 0.f, sy = 0.f, syy = 0.f, sxy = 0.f;
    for (int t = tid; t < T_LEN; t += 256) {
        float xv = x[t], yv = y[t];
        sx += xv; sxx += xv * xv;
        sy += yv; syy += yv * yv;
        sxy += xv * yv;
    }

    __shared__ float red[5][256];
    red[0][tid] = sx; red[1][tid] = sxx; red[2][tid] = sy;
    red[3][tid] = syy; red[4][tid] = sxy;
    __syncthreads();
    for (int s = 128; s > 0; s >>= 1) {
        if (tid < s) {
            #pragma unroll
            for (int j = 0; j < 5; ++j) red[j][tid] += red[j][tid + s];
        }
        __syncthreads();
    }

    __shared__ float bc[2];
    if (tid == 0) {
        const float Tf = (float)T_LEN;
        float SX = red[0][0], SXX = red[1][0], SY = red[2][0],
              SYY = red[3][0], SXY = red[4][0];
        float mean = SX / Tf;
        float var  = (SXX - SX * SX / Tf) / (Tf - 1.0f);   // ddof=1
        float inv  = 1.0f / sqrtf(var);
        // stats of standardized p = (x-mean)*inv (algebraic identities)
        float sum_x  = (SX - Tf * mean) * inv;
        float sum_x2 = (SXX - 2.0f * mean * SX + Tf * mean * mean) * inv * inv;
        float sum_xy = (SXY - mean * SY) * inv;
        float num = Tf * sum_xy - sum_x * SY;
        float den = sqrtf((Tf * sum_x2 - sum_x * sum_x) * (Tf * SYY - SY * SY));
        pearson[row] = num / den;
        bc[0] = mean; bc[1] = inv;
    }
    __syncthreads();

    const float mean = bc[0], inv = bc[1];
    const float hc = 6.28318530717958647692f / (float)(N_TOT - 1);  // hanning
    for (int t = tid; t < T_LEN; t += 256) {
        int n = row * T_LEN + t;
        float h = 0.5f - 0.5f * __cosf(hc * (float)n);
        sig[n] = (x[t] - mean) * inv * h;
    }
}

// ---------------------------------------------------------------------------
// Kernel 2: stage-1 factorized DFT as f32 WMMA GEMM, split-K deterministic.
//   Smat[f,n2] = sum_n1 sin(w_f*64*n1) * sig[n1*64+n2]   (Cmat with cos)
// One wave per (freq-tile, K-segment). EXEC is all 1s (32 threads, no div.).
// ---------------------------------------------------------------------------
__global__ __launch_bounds__(32)
void k_dft_stage1(const float* __restrict__ sig,
                  float* __restrict__ Spart, float* __restrict__ Cpart,
                  const int* __restrict__ frame_rate)
{
    const int ftile = blockIdx.x;          // 0..8
    const int seg   = blockIdx.y;          // 0..KSEG-1
    const int lane  = threadIdx.x;         // wave32
    const int m     = lane & 15;
    const int hi    = lane >> 4;           // 0: K even pair, 1: K odd pair
    const int f     = ftile * 16 + m;      // A-matrix row this lane holds

    // phase step per K unit, in turns, computed in double then reduced mod 1
    const double fr = (double)(*frame_rate);
    double dturn = (40.0 + (double)f) * 64.0 / (60.0 * fr);
    dturn -= floor(dturn);

    v8f accS[4], accC[4];
    #pragma unroll
    for (int nt = 0; nt < 4; ++nt) {
        accS[nt] = (v8f)0.0f;
        accC[nt] = (v8f)0.0f;
    }

    const int kbase0 = seg * KPER;
    for (int kb = 0; kb < KPER; kb += 4) {
        const int K0 = kbase0 + kb + (hi ? 2 : 0);
        // angle in turns, exact range reduction in double
        double t0 = (double)K0 * dturn; t0 -= floor(t0);
        double t1 = t0 + dturn;         if (t1 >= 1.0) t1 -= 1.0;
        float s0, c0, s1, c1;
        __sincosf((float)t0 * 6.28318530717958647692f, &s0, &c0);
        __sincosf((float)t1 * 6.28318530717958647692f, &s1, &c1);
        v2f aS; aS.x = s0; aS.y = s1;   // A[M][K], K = K0, K0+1
        v2f aC; aC.x = c0; aC.y = c1;

        const int krow0 = K0 * N2;      // sig row (n1 index) * 64
        const int krow1 = krow0 + N2;
        #pragma unroll
        for (int nt = 0; nt < 4; ++nt) {
            const int col = nt * 16 + m;           // B lane = column n2
            v2f b; b.x = sig[krow0 + col]; b.y = sig[krow1 + col];
            accS[nt] = __builtin_amdgcn_wmma_f32_16x16x4_f32(
                false, aS, false, b, (short)0, accS[nt], false, false);
            accC[nt] = __builtin_amdgcn_wmma_f32_16x16x4_f32(
                false, aC, false, b, (short)0, accC[nt], false, false);
        }
    }

    // C/D layout: lane&15 = N column, VGPR v -> row v + 8*hi
    #pragma unroll
    for (int nt = 0; nt < 4; ++nt) {
        #pragma unroll
        for (int v = 0; v < 8; ++v) {
            const int row = ftile * 16 + v + 8 * hi;
            const int col = nt * 16 + m;
            const int idx = (seg * NFREQ_PAD + row) * N2 + col;
            Spart[idx] = accS[nt][v];
            Cpart[idx] = accC[nt][v];
        }
    }
}

// ---------------------------------------------------------------------------
// Kernel 3: sum split-K partials
// ---------------------------------------------------------------------------
__global__ __launch_bounds__(256)
void k_reduce_seg(const float* __restrict__ Spart, const float* __restrict__ Cpart,
                  float* __restrict__ Smat, float* __restrict__ Cmat)
{
    int i = blockIdx.x * 256 + threadIdx.x;
    if (i < TILE_ELEMS) {
        float s = 0.f, c = 0.f;
        for (int seg = 0; seg < KSEG; ++seg) {
            s += Spart[seg * TILE_ELEMS + i];
            c += Cpart[seg * TILE_ELEMS + i];
        }
        Smat[i] = s; Cmat[i] = c;
    }
}

// ---------------------------------------------------------------------------
// Kernel 4: stage-2 DFT recombination + all scalar losses (1 block)
// ---------------------------------------------------------------------------
__global__ __launch_bounds__(256)
void k_final(const float* __restrict__ Smat, const float* __restrict__ Cmat,
             const float* __restrict__ pearson,
             const int* __restrict__ avg_hr, const int* __restrict__ frame_rate,
             const int* __restrict__ pa, const int* __restrict__ pb,
             float* __restrict__ out)
{
    __shared__ float ca[NFREQ];
    const int t = threadIdx.x;
    const float fr = (float)(*frame_rate);

    if (t < NFREQ) {
        const float w = 6.28318530717958647692f * (40.0f + (float)t) / (60.0f * fr);
        float sp = 0.f, cp = 0.f;
        for (int n2 = 0; n2 < N2; ++n2) {
            float s, c; __sincosf(w * (float)n2, &s, &c);
            float Cm = Cmat[t * N2 + n2], Sm = Smat[t * N2 + n2];
            sp += s * Cm + c * Sm;   // sin(a+b) recombination
            cp += c * Cm - s * Sm;   // cos(a+b) recombination
        }
        ca[t] = sp * sp + cp * cp;
    }
    __syncthreads();

    if (t == 0) {
        float sum = 0.f;
        for (int i = 0; i < NFREQ; ++i) sum += ca[i];
        for (int i = 0; i < NFREQ; ++i) ca[i] /= sum;

        float mx = -1e30f; int am = 0;
        for (int i = 0; i < NFREQ; ++i)
            if (ca[i] > mx) { mx = ca[i]; am = i; }

        float denom = 0.f;
        for (int i = 0; i < NFREQ; ++i) denom += __expf(ca[i] - mx);
        const float logZ = mx + __logf(denom);          // logsumexp(ca)

        const int tix = (*avg_hr) - 40;
        const float lce = logZ - ca[tix];

        const float INV_SQ2PI = 0.3989422804014327f;    // 1/sqrt(2*pi)
        float kl = 0.f;
        for (int i = 0; i < NFREQ; ++i) {
            float d = (float)i - (float)tix;
            float td = __expf(-0.5f * d * d) * INV_SQ2PI;
            td = fmaxf(td, 1e-15f);
            float log_fd = ca[i] - logZ;                // log softmax
            kl += td * (__logf(td) - log_fd);
        }
        kl /= (float)NFREQ;

        float pr = 0.f;
        for (int r = 0; r < B_ROWS; ++r) pr += 1.0f - pearson[r];
        pr /= (float)B_ROWS;

        const float mae = fabsf((float)tix - (float)am);
        const float total = (float)(*pa) * pr + (float)(*pb) * (lce + kl);
        out[0] = total; out[1] = pr; out[2] = kl; out[3] = lce; out[4] = mae;
    }
}

// ---------------------------------------------------------------------------
extern "C" void kernel_launch(void* const* d_in, const int* in_sizes, int n_in,
                              void* d_out, int out_size, void* d_ws, size_t ws_size,
                              hipStream_t stream)
{
    const float* pred   = (const float*)d_in[0];
    const float* targ   = (const float*)d_in[1];
    const int*   avg_hr = (const int*)d_in[2];
    const int*   frate  = (const int*)d_in[3];
    const int*   pa     = (const int*)d_in[4];
    const int*   pb     = (const int*)d_in[5];

    float* W       = (float*)d_ws;
    float* sig     = W;                              // 262144
    float* Spart   = sig   + N_TOT;                  // KSEG*9216
    float* Cpart   = Spart + KSEG * TILE_ELEMS;      // KSEG*9216
    float* Smat    = Cpart + KSEG * TILE_ELEMS;      // 9216
    float* Cmat    = Smat  + TILE_ELEMS;             // 9216
    float* pearson = Cmat  + TILE_ELEMS;             // 64
    // total: ~2.3 MB of workspace

    k_rowstats<<<B_ROWS, 256, 0, stream>>>(pred, targ, sig, pearson);

    dim3 g1(FTILES, KSEG);
    k_dft_stage1<<<g1, 32, 0, stream>>>(sig, Spart, Cpart, frate);

    k_reduce_seg<<<(TILE_ELEMS + 255) / 256, 256, 0, stream>>>(Spart, Cpart, Smat, Cmat);

    k_final<<<1, 256, 0, stream>>>(Smat, Cmat, pearson, avg_hr, frate, pa, pb,
                                   (float*)d_out);
}